// RoiProposal_41755672051999
// MI455X (gfx1250) — compile-verified
//
#include <hip/hip_runtime.h>
#include <hip/hip_bf16.h>
#include <stdint.h>

// ---------------------------------------------------------------------------
// RPN proposal layer for MI455X (gfx1250, wave32, 320KB LDS/WGP, 192MB L2).
// Pipeline: [decode+histogram] -> [scan/select threshold] -> [compact top-12000]
//           -> [greedy NMS in LDS, async-staged, fused suppress+argmax sweep].
// ---------------------------------------------------------------------------

#define HH        512
#define WW        512
#define NA        9
#define NANCH     (HH * WW * NA)      // 2,359,296
#define TOPN      12000
#define OUTN      2000
#define NMS_THR   0.7f
#define IMMAX     4095.0f
#define MINSZ     16.0f
#define NBINS     65536

// Anchors from _generate_anchors(16, (0.5,1,2), (8,16,32)) — precomputed.
__constant__ float c_anchors[NA][4] = {
    { -84.f,  -40.f,   99.f,   55.f},
    {-176.f,  -88.f,  191.f,  103.f},
    {-360.f, -184.f,  375.f,  199.f},
    { -56.f,  -56.f,   71.f,   71.f},
    {-120.f, -120.f,  135.f,  135.f},
    {-248.f, -248.f,  263.f,  263.f},
    { -36.f,  -80.f,   51.f,   95.f},
    { -80.f, -168.f,   95.f,  183.f},
    {-168.f, -344.f,  183.f,  359.f},
};

// Monotone key: larger float -> larger uint (handles -inf for filtered boxes).
__device__ __forceinline__ uint32_t fkey(float s) {
    uint32_t u = __float_as_uint(s);
    return (u & 0x80000000u) ? ~u : (u | 0x80000000u);
}

__device__ __forceinline__ void decode_one(int i, const float* __restrict__ cls,
                                           const float* __restrict__ bbox,
                                           float4& box, float& score) {
    int a  = i % NA;
    int hw = i / NA;
    int w  = hw % WW;
    int h  = hw / WW;

    // softmax over the 2-class axis == sigmoid(s1 - s0)
    int cbase = hw * (2 * NA);
    float s0 = cls[cbase + a];
    float s1 = cls[cbase + NA + a];
    float sc = 1.0f / (1.0f + expf(s0 - s1));

    float shx = (float)(w * 8);
    float shy = (float)(h * 8);
    float ax1 = c_anchors[a][0] + shx;
    float ay1 = c_anchors[a][1] + shy;
    float ax2 = c_anchors[a][2] + shx;
    float ay2 = c_anchors[a][3] + shy;

    float ws = ax2 - ax1 + 1.0f;
    float hs = ay2 - ay1 + 1.0f;
    float cx = ax1 + 0.5f * ws;
    float cy = ay1 + 0.5f * hs;

    const float4 d = *(const float4*)(bbox + (size_t)i * 4);  // dx,dy,dw,dh
    float pcx = d.x * ws + cx;
    float pcy = d.y * hs + cy;
    float pw  = expf(d.z) * ws;
    float ph  = expf(d.w) * hs;

    float x1 = fminf(fmaxf(pcx - 0.5f * pw, 0.0f), IMMAX);
    float y1 = fminf(fmaxf(pcy - 0.5f * ph, 0.0f), IMMAX);
    float x2 = fminf(fmaxf(pcx + 0.5f * pw, 0.0f), IMMAX);
    float y2 = fminf(fmaxf(pcy + 0.5f * ph, 0.0f), IMMAX);

    bool keep = ((x2 - x1 + 1.0f) >= MINSZ) && ((y2 - y1 + 1.0f) >= MINSZ);
    box   = make_float4(x1, y1, x2, y2);
    score = keep ? sc : -__builtin_inff();
}

// ---- Pass 1: decode + 16-bit-key histogram (global atomics, L2-resident) ----
__global__ void k_decode_hist(const float* __restrict__ cls,
                              const float* __restrict__ bbox,
                              uint32_t* __restrict__ hist) {
    int i = blockIdx.x * blockDim.x + threadIdx.x;
    if (i >= NANCH) return;
    float4 box; float s;
    decode_one(i, cls, bbox, box, s);
    atomicAdd(&hist[fkey(s) >> 16], 1u);
}

// ---- Pass 2: one block. Suffix-scan 65536 bins, find threshold bin such that
//      count(bin >= thresh) >= TOPN > count(bin > thresh). Also init outputs. --
__global__ __launch_bounds__(1024) void k_scan_select(
        const uint32_t* __restrict__ hist, int* __restrict__ meta,
        float* __restrict__ topScore, float4* __restrict__ topBox) {
    __shared__ uint32_t csum[1024];
    int t = threadIdx.x;

    // Defensive init of the top-K arrays (slots may stay unfilled if <TOPN
    // boxes survive min-size filtering; NMS tail then matches the reference).
    for (int j = t; j < TOPN; j += 1024) {
        topScore[j] = -__builtin_inff();
        topBox[j]   = make_float4(0.f, 0.f, 0.f, 0.f);
    }

    const uint32_t base = (uint32_t)t * 64u;
    uint32_t s = 0;
    #pragma unroll 4
    for (int b = 0; b < 64; ++b) s += hist[base + b];
    csum[t] = s;
    __syncthreads();
    // inclusive suffix scan over the 1024 chunk sums
    for (int off = 1; off < 1024; off <<= 1) {
        uint32_t v = csum[t];
        if (t + off < 1024) v += csum[t + off];
        __syncthreads();
        csum[t] = v;
        __syncthreads();
    }
    uint32_t c = (t + 1 < 1024) ? csum[t + 1] : 0u;  // count of bins in chunks > t
    for (int b = 63; b >= 0; --b) {
        uint32_t h    = hist[base + (uint32_t)b];
        uint32_t cinc = c + h;                        // C(bin) = count(bin' >= bin)
        if (cinc >= TOPN && c < TOPN) {               // exactly one bin matches
            meta[0] = (int)(base + (uint32_t)b);      // threshold bin
            meta[1] = (int)c;                         // count strictly above it
        }
        c = cinc;
    }
}

// ---- Pass 3: re-decode (hits 192MB L2), compact top-TOPN into workspace -----
__global__ void k_compact(const float* __restrict__ cls,
                          const float* __restrict__ bbox,
                          const int* __restrict__ meta, int* __restrict__ cnt,
                          float* __restrict__ topScore, float4* __restrict__ topBox) {
    int i = blockIdx.x * blockDim.x + threadIdx.x;
    if (i >= NANCH) return;
    float4 box; float s;
    decode_one(i, cls, bbox, box, s);
    int bin    = (int)(fkey(s) >> 16);
    int thresh = meta[0];
    int above  = meta[1];
    if (bin > thresh) {
        int pos = atomicAdd(&cnt[0], 1);
        if (pos < TOPN) { topScore[pos] = s; topBox[pos] = box; }
    } else if (bin == thresh) {
        int pos = above + atomicAdd(&cnt[1], 1);
        if (pos < TOPN) { topScore[pos] = s; topBox[pos] = box; }
    }
}

// ---- Pass 4: greedy NMS, single 1024-thread workgroup (32 wave32s).
//      Whole working set (240KB) lives in LDS — unique CDNA5 capability.
//      Staged with async global->LDS b128 copies (ASYNCcnt path).
//      Per round: ONE fused sweep (suppress vs winner + next argmax) + 2 barriers.
__global__ __launch_bounds__(1024) void k_nms(
        const float4* __restrict__ topBox, const float* __restrict__ topScore,
        float* __restrict__ out) {
    __shared__ float4 s_box[TOPN];     // 192,000 B
    __shared__ float  s_sc[TOPN];      //  48,000 B
    __shared__ float  r_val[32];
    __shared__ int    r_idx[32];
    __shared__ int    s_widx;

    const int t = threadIdx.x;
    const uint32_t lds_box = (uint32_t)(uintptr_t)(void*)&s_box[0];
    const uint32_t lds_sc  = (uint32_t)(uintptr_t)(void*)&s_sc[0];

    // Async-stage boxes: 12000 x 16B, one b128 per lane per iteration.
    for (int j = t; j < TOPN; j += 1024) {
        asm volatile("global_load_async_to_lds_b128 %0, %1, off"
                     :: "v"(lds_box + (uint32_t)j * 16u), "v"(topBox + j)
                     : "memory");
    }
    // Async-stage scores: 3000 x 16B chunks.
    for (int j = t; j < TOPN / 4; j += 1024) {
        asm volatile("global_load_async_to_lds_b128 %0, %1, off"
                     :: "v"(lds_sc + (uint32_t)j * 16u), "v"(topScore + j * 4)
                     : "memory");
    }
    asm volatile("s_wait_asynccnt 0" ::: "memory");
    __syncthreads();

    const float NEG_INF = -__builtin_inff();

    // ---- initial argmax (ties -> lowest index, matching jnp.argmax) ----
    {
        float bv = s_sc[t];
        int   bi = t;
        for (int j = t + 1024; j < TOPN; j += 1024) {
            float v = s_sc[j];
            if (v > bv) { bv = v; bi = j; }
        }
        #pragma unroll
        for (int m = 16; m > 0; m >>= 1) {
            float ov = __shfl_xor(bv, m, 32);
            int   oi = __shfl_xor(bi, m, 32);
            if (ov > bv || (ov == bv && oi < bi)) { bv = ov; bi = oi; }
        }
        if ((t & 31) == 0) { r_val[t >> 5] = bv; r_idx[t >> 5] = bi; }
        __syncthreads();
        if (t < 32) {
            bv = r_val[t]; bi = r_idx[t];
            #pragma unroll
            for (int m = 16; m > 0; m >>= 1) {
                float ov = __shfl_xor(bv, m, 32);
                int   oi = __shfl_xor(bi, m, 32);
                if (ov > bv || (ov == bv && oi < bi)) { bv = ov; bi = oi; }
            }
            if (t == 0) s_widx = bi;
        }
        __syncthreads();
    }

    for (int i = 0; i < OUTN; ++i) {
        const int    widx  = s_widx;
        const float4 wb    = s_box[widx];          // boxes never mutate
        const float  warea = (wb.z - wb.x + 1.0f) * (wb.w - wb.y + 1.0f);

        if (t == 0) {
            out[i * 5 + 0] = 0.0f;
            out[i * 5 + 1] = wb.x;
            out[i * 5 + 2] = wb.y;
            out[i * 5 + 3] = wb.z;
            out[i * 5 + 4] = wb.w;
        }

        // ---- fused sweep: suppress vs winner AND find next argmax ----
        // All-suppressed tail: bv stays -inf, bi stays 0 -> winner 0, matching
        // jnp.argmax over an all -inf vector.
        float bv = NEG_INF;
        int   bi = 0;
        for (int j = t; j < TOPN; j += 1024) {
            float v = s_sc[j];
            if (v == NEG_INF) continue;            // already out: skip box read
            float4 b  = s_box[j];
            float xx1 = fmaxf(wb.x, b.x);
            float yy1 = fmaxf(wb.y, b.y);
            float xx2 = fminf(wb.z, b.z);
            float yy2 = fminf(wb.w, b.w);
            float iw  = fmaxf(0.0f, xx2 - xx1 + 1.0f);
            float ih  = fmaxf(0.0f, yy2 - yy1 + 1.0f);
            float inter = iw * ih;
            float area  = (b.z - b.x + 1.0f) * (b.w - b.y + 1.0f);
            float iou   = inter / (warea + area - inter);
            if (iou > NMS_THR) {
                s_sc[j] = NEG_INF;                 // includes the winner itself
            } else if (v > bv || (v == bv && j < bi)) {
                bv = v; bi = j;
            }
        }
        #pragma unroll
        for (int m = 16; m > 0; m >>= 1) {
            float ov = __shfl_xor(bv, m, 32);
            int   oi = __shfl_xor(bi, m, 32);
            if (ov > bv || (ov == bv && oi < bi)) { bv = ov; bi = oi; }
        }
        if ((t & 31) == 0) { r_val[t >> 5] = bv; r_idx[t >> 5] = bi; }
        __syncthreads();
        if (t < 32) {
            bv = r_val[t]; bi = r_idx[t];
            #pragma unroll
            for (int m = 16; m > 0; m >>= 1) {
                float ov = __shfl_xor(bv, m, 32);
                int   oi = __shfl_xor(bi, m, 32);
                if (ov > bv || (ov == bv && oi < bi)) { bv = ov; bi = oi; }
            }
            if (t == 0) s_widx = bi;
        }
        __syncthreads();
    }
}

extern "C" void kernel_launch(void* const* d_in, const int* in_sizes, int n_in,
                              void* d_out, int out_size, void* d_ws, size_t ws_size,
                              hipStream_t stream) {
    const float* cls  = (const float*)d_in[0];   // (1,512,512,18) f32
    const float* bbox = (const float*)d_in[1];   // (1,512,512,36) f32
    float* out = (float*)d_out;                  // (2000,5) f32

    // Workspace layout (~0.5 MB total):
    //   [0, 262144)        : uint32 hist[65536]
    //   [262144, 262208)   : int meta[16]  (meta[0]=thresh bin, meta[1]=count above,
    //                                       meta[2..3]=compaction counters)
    //   [262208, 310208)   : float topScore[12000]
    //   [310208, 502208)   : float4 topBox[12000]
    uint8_t*  ws       = (uint8_t*)d_ws;
    uint32_t* hist     = (uint32_t*)ws;
    int*      meta     = (int*)(ws + 262144);
    float*    topScore = (float*)(ws + 262208);
    float4*   topBox   = (float4*)(ws + 310208);

    // Zero hist + meta every call (atomics accumulate; must be deterministic).
    hipMemsetAsync(ws, 0, 262208, stream);

    const int blocks = (NANCH + 255) / 256;
    k_decode_hist<<<blocks, 256, 0, stream>>>(cls, bbox, hist);
    k_scan_select<<<1, 1024, 0, stream>>>(hist, meta, topScore, topBox);
    k_compact<<<blocks, 256, 0, stream>>>(cls, bbox, meta, meta + 2, topScore, topBox);
    k_nms<<<1, 1024, 0, stream>>>(topBox, topScore, out);
}